// SinkhornLayer_66099546685706
// MI455X (gfx1250) — compile-verified
//
#include <hip/hip_runtime.h>
#include <stdint.h>

// Sinkhorn (dual-variable, linear-domain) for S:(256,192,256) f32, pad to 256x256,
// 20 iterations, output exp(log_alpha) : (256,256,256) f32.
//
// Math: with A = S/tau (zero dummy rows), the reference iteration
//   log_alpha -= rowLSE; log_alpha -= colLSE
// is exactly  u[r]=LSE_c(A-v), v[c]=LSE_r(A-u)  with matrix = A-u-v.
// Linear domain: K=exp(A), bu=exp(-u), bv=exp(-v):
//   bu[r] = 1 / sum_c K[r,c]*bv[c]
//   bv[c] = 1 / ( sum_{r<192} K[r,c]*bu[r] + 64*bu_d ),  bu_d = 1/sum_c bv[c]
// Output: K*bu*bv (dummy rows: bu_d*bv[c]).  K stays LDS-resident; TDM stages it
// with hardware row padding to a 260-dword stride so every access is a b128 at
// the ideal 2-clk/wave LDS service rate.

#define N_BATCH 256
#define NL 192
#define NH 256
#define RSTRIDE 260        // padded row stride (dwords): 16B multiple, b128-ideal banks
#define N_ITER 20

typedef uint32_t u32x4 __attribute__((ext_vector_type(4)));
typedef int      i32x4 __attribute__((ext_vector_type(4)));
typedef int      i32x8 __attribute__((ext_vector_type(8)));
typedef float    f32x4 __attribute__((ext_vector_type(4)));

// LDS layout (dword offsets)
#define TILE_OFF  0
#define BU_OFF    (NL * RSTRIDE)          // 49920: bu[0..191], bu[192]=dummy scale
#define BV_OFF    (BU_OFF + 256)          // 50176
#define UPART_OFF (BV_OFF + 256)          // 50432: 8 groups x 192 rows
#define VPART_OFF (UPART_OFF + 8 * NL)    // 51968: 4 bands x 256 cols
#define SMEM_DW   (VPART_OFF + 4 * NH)    // 52992 dwords = 211968 B  (< 320KB WGP LDS)

#if defined(__AMDGCN__) && __has_builtin(__builtin_amdgcn_tensor_load_to_lds)
#define HAVE_TDM 1
#else
#define HAVE_TDM 0
#endif

__global__ __launch_bounds__(256) void sinkhorn_cdna5(const float* __restrict__ S,
                                                      const int* __restrict__ tau_p,
                                                      float* __restrict__ out) {
  extern __shared__ float smem[];
  float* tile  = smem + TILE_OFF;
  float* bu    = smem + BU_OFF;
  float* bv    = smem + BV_OFF;
  float* upart = smem + UPART_OFF;
  float* vpart = smem + VPART_OFF;

  const int t = threadIdx.x;             // 0..255
  const int g = t >> 5;                  // wave id 0..7
  const int i = t & 31;                  // lane
  const int band  = g >> 1;              // 0..3 row band
  const int chalf = (g & 1) * 128;       // column half
  const int c0 = chalf + 4 * i;          // this thread's 4 columns
  const int b = blockIdx.x;
  const float inv_tau = 1.0f / (float)tau_p[0];
  const float* Sb = S + (size_t)b * (NL * NH);

#if HAVE_TDM
  // --- Stage raw S tile via Tensor Data Mover; hardware pads each 256-dword row
  //     by 4 dwords (pad_interval=7 -> 256dw, pad_amount=3 -> 4dw) => stride 260.
  if (t < 32) {
    uint64_t ga = (uint64_t)(uintptr_t)Sb;
    uint32_t lds_base = (uint32_t)(uintptr_t)tile;
    u32x4 g0 = { 1u, lds_base, (uint32_t)ga,
                 (uint32_t)((ga >> 32) & 0x01FFFFFFull) | (2u << 30) };
    i32x8 g1 = { (int)((2u << 16) | (1u << 20) | (7u << 22) | (3u << 25)),
                 (int)(((uint32_t)NH & 0xFFFFu) << 16),   // tensor_dim0 lo16<<16
                 (int)(((uint32_t)NL & 0xFFFFu) << 16),   // tensor_dim1 lo16<<16
                 (int)(((uint32_t)NH & 0xFFFFu) << 16),   // tile_dim0<<16
                 (int)NL,                                 // tile_dim1
                 (int)NH,                                 // tensor_dim0_stride lo32
                 0, 0 };
    i32x4 gz4 = { 0, 0, 0, 0 };
#if __has_include(<hip/amd_detail/amd_gfx1250_TDM.h>)
    i32x8 gz8 = { 0, 0, 0, 0, 0, 0, 0, 0 };
    __builtin_amdgcn_tensor_load_to_lds(g0, g1, gz4, gz4, gz8, 0);
#else
    __builtin_amdgcn_tensor_load_to_lds(g0, g1, gz4, gz4, 0);
#endif
#if __has_builtin(__builtin_amdgcn_s_wait_tensorcnt)
    __builtin_amdgcn_s_wait_tensorcnt(0);
#else
    asm volatile("s_wait_tensorcnt 0x0" ::: "memory");
#endif
  }
  bv[t] = 1.0f;
  __syncthreads();
  // In-place K = exp(S/tau), 4-wide (48 rows x 4 cols per thread)
  #pragma unroll 4
  for (int rr = 0; rr < 48; ++rr) {
    const int r = 48 * band + rr;
    f32x4* p = (f32x4*)(tile + r * RSTRIDE + c0);
    f32x4 x = *p;
    f32x4 e = { __expf(x.x * inv_tau), __expf(x.y * inv_tau),
                __expf(x.z * inv_tau), __expf(x.w * inv_tau) };
    *p = e;
  }
#else
  // Fallback: coalesced b128 global loads, exp on the fly
  bv[t] = 1.0f;
  #pragma unroll 4
  for (int rr = 0; rr < 48; ++rr) {
    const int r = 48 * band + rr;
    f32x4 x = *(const f32x4*)(Sb + r * NH + c0);
    f32x4 e = { __expf(x.x * inv_tau), __expf(x.y * inv_tau),
                __expf(x.z * inv_tau), __expf(x.w * inv_tau) };
    *(f32x4*)(tile + r * RSTRIDE + c0) = e;
  }
#endif
  __syncthreads();

  for (int it = 0; it < N_ITER; ++it) {
    // ---- u-step phase A: group g = 32-column chunk, bv chunk in registers,
    //      each thread does 6 interleaved rows (banks 4*lane: b128-ideal)
    {
      f32x4 w[8];
      #pragma unroll
      for (int q = 0; q < 8; ++q) w[q] = *(const f32x4*)(bv + 32 * g + 4 * q);
      #pragma unroll
      for (int k = 0; k < 6; ++k) {
        const int r = i + 32 * k;
        const f32x4* row = (const f32x4*)(tile + r * RSTRIDE + 32 * g);
        f32x4 a = { 0.f, 0.f, 0.f, 0.f };
        #pragma unroll
        for (int q = 0; q < 8; ++q) a += row[q] * w[q];
        upart[g * NL + r] = a.x + a.y + a.z + a.w;
      }
    }
    __syncthreads();
    // ---- u-step phase B: combine 8 group partials; threads >=192 compute bu_d
    if (t < NL) {
      float s = 0.f;
      #pragma unroll
      for (int q = 0; q < 8; ++q) s += upart[q * NL + t];
      bu[t] = 1.0f / s;
    } else {
      float s = 0.f;
      #pragma unroll 16
      for (int c4 = 0; c4 < 64; ++c4) {
        f32x4 x = ((const f32x4*)bv)[c4];
        s += x.x + x.y + x.z + x.w;
      }
      bu[t] = 1.0f / s;                 // only bu[NL] is consumed (dummy rows)
    }
    __syncthreads();

    // ---- v-step phase A: band = 48 rows, thread owns 4 columns; bu via b128
    {
      f32x4 acc = { 0.f, 0.f, 0.f, 0.f };
      #pragma unroll
      for (int rr = 0; rr < 12; ++rr) {
        const int r = 48 * band + 4 * rr;
        f32x4 w4 = *(const f32x4*)(bu + r);
        const float* base = tile + r * RSTRIDE + c0;
        acc += *(const f32x4*)(base)               * w4.x;
        acc += *(const f32x4*)(base + RSTRIDE)     * w4.y;
        acc += *(const f32x4*)(base + 2 * RSTRIDE) * w4.z;
        acc += *(const f32x4*)(base + 3 * RSTRIDE) * w4.w;
      }
      *(f32x4*)(vpart + band * NH + c0) = acc;
    }
    __syncthreads();
    // ---- v-step phase B: combine 4 band partials + dummy-row mass
    {
      float s = (float)(NH - NL) * bu[NL];
      #pragma unroll
      for (int q = 0; q < 4; ++q) s += vpart[q * NH + t];
      bv[t] = 1.0f / s;
    }
    __syncthreads();
  }

  // ---- Output: exp(log_alpha) = K*bu*bv ; dummy rows = bu_d*bv.
  //      Thread: 64 rows x its 4 columns; wave stores contiguous 512B b128.
  const f32x4 wv = *(const f32x4*)(bv + c0);
  const float bud = bu[NL];
  float* outb = out + (size_t)b * (NH * NH);
  #pragma unroll 8
  for (int rr = 0; rr < 64; ++rr) {
    const int r = 64 * band + rr;                // bands 0..3 cover rows 0..255
    f32x4 val;
    if (r < NL) {                                 // wave-uniform branch
      f32x4 kv = *(const f32x4*)(tile + r * RSTRIDE + c0);
      val = kv * wv * bu[r];
    } else {
      val = wv * bud;
    }
    *(f32x4*)(outb + r * NH + c0) = val;
  }
}

extern "C" void kernel_launch(void* const* d_in, const int* in_sizes, int n_in,
                              void* d_out, int out_size, void* d_ws, size_t ws_size,
                              hipStream_t stream) {
  (void)in_sizes; (void)n_in; (void)out_size; (void)d_ws; (void)ws_size;
  const float* S = (const float*)d_in[0];
  // d_in[1]=num_logical(192), d_in[2]=num_physical(256) are hardcoded.
  const int* tau = (const int*)d_in[3];
  float* out = (float*)d_out;

  const size_t shmem = (size_t)SMEM_DW * sizeof(float);   // 211,968 B
  sinkhorn_cdna5<<<N_BATCH, 256, shmem, stream>>>(S, tau, out);
}